// EEGGENET_33956011442392
// MI455X (gfx1250) — compile-verified
//
#include <hip/hip_runtime.h>
#include <math.h>

typedef float v2f __attribute__((ext_vector_type(2)));
typedef float v8f __attribute__((ext_vector_type(8)));

#define EPS_BN 1e-3f

// ---------------------------------------------------------------------------
// Setup: build normalized adjacencies, renorm spatial weights, fold the graph
// conv + spatial depthwise conv into weff[48][64], fold BN1/BN2 into per-oc
// scale/bias, fold BN3, renorm classifier weights.
// ---------------------------------------------------------------------------
__global__ __launch_bounds__(256) void eeg_setup_kernel(
    const float* __restrict__ g1, const float* __restrict__ b1,
    const float* __restrict__ m1, const float* __restrict__ v1,
    const float* __restrict__ e1, const float* __restrict__ e2,
    const float* __restrict__ sw, const float* __restrict__ g2,
    const float* __restrict__ b2, const float* __restrict__ m2,
    const float* __restrict__ v2, const float* __restrict__ g3,
    const float* __restrict__ b3, const float* __restrict__ m3,
    const float* __restrict__ v3, const float* __restrict__ clw,
    float* __restrict__ weff, float* __restrict__ S, float* __restrict__ Bi,
    float* __restrict__ a3o, float* __restrict__ c3o, float* __restrict__ cwr)
{
    __shared__ float A1n[64][64];
    __shared__ float A2n[64][64];
    __shared__ float scl[48];
    __shared__ float wsum[48];
    __shared__ float weff_s[48 * 64];
    const int tid = threadIdx.x;

    // Build row-normalized adjacencies A = tril(e)+tril(e)^T+I, row-normalize.
    if (tid < 64) {
        const int i = tid;
        float s1 = 0.f, s2 = 0.f;
        for (int j = 0; j < 64; ++j) {
            float x1, x2;
            if (i == j)      { x1 = 1.f; x2 = 1.f; }
            else if (i > j)  { int k = i * (i - 1) / 2 + j; x1 = e1[k]; x2 = e2[k]; }
            else             { int k = j * (j - 1) / 2 + i; x1 = e1[k]; x2 = e2[k]; }
            s1 += x1; s2 += x2;
        }
        const float inv1 = 1.f / s1, inv2 = 1.f / s2;
        for (int j = 0; j < 64; ++j) {
            float x1, x2;
            if (i == j)      { x1 = 1.f; x2 = 1.f; }
            else if (i > j)  { int k = i * (i - 1) / 2 + j; x1 = e1[k]; x2 = e2[k]; }
            else             { int k = j * (j - 1) / 2 + i; x1 = e1[k]; x2 = e2[k]; }
            A1n[i][j] = x1 * inv1;
            A2n[i][j] = x2 * inv2;
        }
    }
    __syncthreads();

    // Max-norm(1.0) renorm scale for the spatial depthwise weights (48 x 64).
    if (tid < 48) {
        float ss = 0.f;
        for (int j = 0; j < 64; ++j) { float w = sw[tid * 64 + j]; ss += w * w; }
        float n = sqrtf(ss);
        scl[tid] = fminf(1.0f, 1.0f / fmaxf(n, 1e-7f));
    }
    __syncthreads();

    // weff[oc][j]: hop0 -> swr[oc][j]; hop1/2 -> (A^T swr[oc])[j].
    for (int idx = tid; idx < 48 * 64; idx += blockDim.x) {
        const int oc = idx >> 6, j = idx & 63;
        const int hop = oc >> 4;   // oc/2 gives group (0..23); group/8 = hop
        const float sc = scl[oc];
        float val;
        if (hop == 0) {
            val = sw[oc * 64 + j] * sc;
        } else {
            float acc = 0.f;
            if (hop == 1) { for (int c = 0; c < 64; ++c) acc = fmaf(A1n[c][j], sw[oc * 64 + c], acc); }
            else          { for (int c = 0; c < 64; ++c) acc = fmaf(A2n[c][j], sw[oc * 64 + c], acc); }
            val = acc * sc;
        }
        weff_s[idx] = val;
        weff[idx]   = val;
    }
    __syncthreads();

    if (tid < 48) {
        float s = 0.f;
        for (int j = 0; j < 64; ++j) s += weff_s[tid * 64 + j];
        wsum[tid] = s;
    }
    __syncthreads();

    // Fold BN1 (per f) and BN2 (per oc) into one affine per oc applied after
    // the temporal conv on z:  s = S[oc]*conv + Bi[oc].
    if (tid < 48) {
        const int oc = tid;
        const int f  = (oc >> 1) & 7;
        const float A1f = g1[f] * rsqrtf(v1[f] + EPS_BN);
        const float C1f = b1[f] - m1[f] * A1f;
        const float A2c = g2[oc] * rsqrtf(v2[oc] + EPS_BN);
        const float C2c = b2[oc] - m2[oc] * A2c;
        S[oc]  = A2c * A1f;
        Bi[oc] = A2c * C1f * wsum[oc] + C2c;
    }
    if (tid < 16) {
        const float a = g3[tid] * rsqrtf(v3[tid] + EPS_BN);
        a3o[tid] = a;
        c3o[tid] = b3[tid] - m3[tid] * a;
    }
    __syncthreads();

    // Max-norm(0.25) renorm of classifier weights (4 x 240).
    if (tid < 4) {
        float ss = 0.f;
        for (int k = 0; k < 240; ++k) { float w = clw[tid * 240 + k]; ss += w * w; }
        const float sc = fminf(1.0f, 0.25f / fmaxf(sqrtf(ss), 1e-7f));
        for (int k = 0; k < 240; ++k) cwr[tid * 240 + k] = clw[tid * 240 + k] * sc;
    }
}

// ---------------------------------------------------------------------------
// WMMA GEMM:  z[b, 0:48, u0:u0+16] = weff(48x64) @ x[b](64x1000) tile.
// One wave per block; EXEC stays all-ones (no divergent guards anywhere).
// GEMM columns are independent, so lanes whose output column falls past 1000
// (last tile only) just load a clamped in-bounds address; their accumulators
// are garbage but never stored.  This keeps the inner loop branch-free:
// 5 unconditional loads + 3 WMMAs per k-step.
// A fragment (16x4 f32): lanes 0-15 hold M=lane, VGPR0/1 = K,K+1; lanes 16-31
// hold K+2,K+3.  B fragment (4x16): lane = N column, same K split.  D: 8 VGPRs,
// rows 0-7 (lanes 0-15) / 8-15 (lanes 16-31), col = lane&15.
// ---------------------------------------------------------------------------
__global__ __launch_bounds__(32) void eeg_gemm_z_kernel(
    const float* __restrict__ x, const float* __restrict__ weff,
    float* __restrict__ z)
{
    const int lane = threadIdx.x;
    const int u0   = blockIdx.x * 16;
    const int b    = blockIdx.y;
    const float* xb = x + (size_t)b * 64 * 1000;

    v8f c0 = {}, c1 = {}, c2 = {};
    const int mrow = lane & 15;
    const int koff = (lane >> 4) << 1;
    const int ucol = u0 + (lane & 15);
    const bool uok = (ucol < 1000);
    const int uc   = uok ? ucol : 999;   // clamped, always in-bounds

    for (int k0 = 0; k0 < 64; k0 += 4) {
        const int k = k0 + koff;
        v2f bf;
        bf[0] = xb[k * 1000 + uc];
        bf[1] = xb[(k + 1) * 1000 + uc];
        v2f a0, a1, a2;
        a0[0] = weff[mrow * 64 + k];        a0[1] = weff[mrow * 64 + k + 1];
        a1[0] = weff[(16 + mrow) * 64 + k]; a1[1] = weff[(16 + mrow) * 64 + k + 1];
        a2[0] = weff[(32 + mrow) * 64 + k]; a2[1] = weff[(32 + mrow) * 64 + k + 1];
        c0 = __builtin_amdgcn_wmma_f32_16x16x4_f32(false, a0, false, bf, (short)0, c0, false, false);
        c1 = __builtin_amdgcn_wmma_f32_16x16x4_f32(false, a1, false, bf, (short)0, c1, false, false);
        c2 = __builtin_amdgcn_wmma_f32_16x16x4_f32(false, a2, false, bf, (short)0, c2, false, false);
    }

    if (uok) {
        const int rbase = (lane < 16) ? 0 : 8;
        float* zb = z + (size_t)b * 48 * 1000;
        #pragma unroll
        for (int r = 0; r < 8; ++r) {
            zb[(rbase + r) * 1000 + ucol]      = c0[r];
            zb[(16 + rbase + r) * 1000 + ucol] = c1[r];
            zb[(32 + rbase + r) * 1000 + ucol] = c2[r];
        }
    }
}

// ---------------------------------------------------------------------------
// 64-tap temporal conv on z (zero-padded 32 each side) + folded BN affine +
// ELU + avg-pool8 -> p1[b, oc, 0:125].  One block per (oc, b).
// ---------------------------------------------------------------------------
__global__ __launch_bounds__(256) void eeg_temporal_kernel(
    const float* __restrict__ z, const float* __restrict__ tw,
    const float* __restrict__ S, const float* __restrict__ Bi,
    float* __restrict__ p1)
{
    __shared__ float zs[1064];
    __shared__ float tps[64];
    __shared__ float selu[1000];
    const int oc = blockIdx.x;
    const int b  = blockIdx.y;
    const int f  = (oc >> 1) & 7;
    const int tid = threadIdx.x;
    const float* zrow = z + ((size_t)b * 48 + oc) * 1000;

    for (int i = tid; i < 1064; i += 256) {
        const int u = i - 32;
        zs[i] = (u >= 0 && u < 1000) ? zrow[u] : 0.f;
    }
    if (tid < 64) tps[tid] = tw[f * 64 + tid];
    __syncthreads();

    const float sc = S[oc], bi = Bi[oc];
    for (int t = tid; t < 1000; t += 256) {
        float acc = 0.f;
        #pragma unroll 8
        for (int k = 0; k < 64; ++k) acc = fmaf(tps[k], zs[t + k], acc);
        const float v = sc * acc + bi;
        selu[t] = (v > 0.f) ? v : expm1f(v);
    }
    __syncthreads();

    if (tid < 125) {
        float s = 0.f;
        #pragma unroll
        for (int k = 0; k < 8; ++k) s += selu[tid * 8 + k];
        p1[((size_t)b * 48 + oc) * 125 + tid] = s * 0.125f;
    }
}

// ---------------------------------------------------------------------------
// Head: depthwise 16-tap conv (pad 8) -> pointwise 48->16 -> BN3+ELU ->
// pool8 -> renormed 15-tap classifier.  One block per batch sample, all LDS.
// ---------------------------------------------------------------------------
__global__ __launch_bounds__(128) void eeg_head_kernel(
    const float* __restrict__ p1, const float* __restrict__ dw,
    const float* __restrict__ pw, const float* __restrict__ a3,
    const float* __restrict__ c3, const float* __restrict__ cwr,
    const float* __restrict__ cb, float* __restrict__ out)
{
    __shared__ float p1s[48 * 125];
    __shared__ float qs[48 * 126];
    __shared__ float r2[16 * 126];
    __shared__ float pm[16 * 15];
    const int b = blockIdx.x;
    const int tid = threadIdx.x;
    const float* pb = p1 + (size_t)b * 48 * 125;

    for (int i = tid; i < 48 * 125; i += 128) p1s[i] = pb[i];
    __syncthreads();

    for (int i = tid; i < 48 * 126; i += 128) {
        const int oc = i / 126, t = i % 126;
        float acc = 0.f;
        #pragma unroll
        for (int k = 0; k < 16; ++k) {
            const int u = t + k - 8;
            if (u >= 0 && u < 125) acc = fmaf(dw[oc * 16 + k], p1s[oc * 125 + u], acc);
        }
        qs[i] = acc;
    }
    __syncthreads();

    for (int i = tid; i < 16 * 126; i += 128) {
        const int m = i / 126, t = i % 126;
        float acc = 0.f;
        for (int oc = 0; oc < 48; ++oc) acc = fmaf(pw[m * 48 + oc], qs[oc * 126 + t], acc);
        const float v = a3[m] * acc + c3[m];
        r2[i] = (v > 0.f) ? v : expm1f(v);
    }
    __syncthreads();

    for (int i = tid; i < 240; i += 128) {
        const int m = i / 15, tp = i % 15;
        float s = 0.f;
        #pragma unroll
        for (int k = 0; k < 8; ++k) s += r2[m * 126 + tp * 8 + k];
        pm[i] = s * 0.125f;
    }
    __syncthreads();

    if (tid < 4) {
        float acc = cb[tid];
        for (int j = 0; j < 240; ++j) acc = fmaf(cwr[tid * 240 + j], pm[j], acc);
        out[b * 4 + tid] = acc;
    }
}

// ---------------------------------------------------------------------------
extern "C" void kernel_launch(void* const* d_in, const int* in_sizes, int n_in,
                              void* d_out, int out_size, void* d_ws, size_t ws_size,
                              hipStream_t stream) {
    const float* x   = (const float*)d_in[0];
    const float* tw  = (const float*)d_in[1];
    const float* g1  = (const float*)d_in[2];
    const float* b1  = (const float*)d_in[3];
    const float* m1  = (const float*)d_in[4];
    const float* v1  = (const float*)d_in[5];
    const float* e1  = (const float*)d_in[6];
    const float* e2  = (const float*)d_in[7];
    const float* sw  = (const float*)d_in[8];
    const float* g2  = (const float*)d_in[9];
    const float* b2  = (const float*)d_in[10];
    const float* m2  = (const float*)d_in[11];
    const float* v2  = (const float*)d_in[12];
    const float* dw  = (const float*)d_in[13];
    const float* pw  = (const float*)d_in[14];
    const float* g3  = (const float*)d_in[15];
    const float* b3  = (const float*)d_in[16];
    const float* m3  = (const float*)d_in[17];
    const float* v3  = (const float*)d_in[18];
    const float* clw = (const float*)d_in[19];
    const float* cb  = (const float*)d_in[20];

    float* ws   = (float*)d_ws;
    float* z    = ws;                 // 64*48*1000 = 3,072,000
    float* p1   = ws + 3072000;       // 64*48*125  =   384,000
    float* weff = ws + 3456000;       // 48*64      =     3,072
    float* S    = ws + 3459072;       // 48
    float* Bi   = ws + 3459120;       // 48
    float* a3   = ws + 3459168;       // 16
    float* c3   = ws + 3459184;       // 16
    float* cwr  = ws + 3459200;       // 4*240 = 960  (end: 3,460,160 floats)

    eeg_setup_kernel<<<1, 256, 0, stream>>>(g1, b1, m1, v1, e1, e2, sw, g2, b2,
                                            m2, v2, g3, b3, m3, v3, clw,
                                            weff, S, Bi, a3, c3, cwr);
    eeg_gemm_z_kernel<<<dim3(63, 64), 32, 0, stream>>>(x, weff, z);
    eeg_temporal_kernel<<<dim3(48, 64), 256, 0, stream>>>(z, tw, S, Bi, p1);
    eeg_head_kernel<<<64, 128, 0, stream>>>(p1, dw, pw, a3, c3, cwr, cb,
                                            (float*)d_out);
}